// RKDAngleLoss_57019985822344
// MI455X (gfx1250) — compile-verified
//
#include <hip/hip_runtime.h>
#include <math.h>

#define BB 256
#define CC 1000
#define TEMP_INV 0.25f
#define EPSN 1e-12f
#define LAMBDA_ 100.0f

typedef __attribute__((ext_vector_type(2))) float v2f;
typedef __attribute__((ext_vector_type(8))) float v8f;

// ---------------- block reduction helpers (256 threads, deterministic) ----------------
__device__ __forceinline__ float blk_reduce_max(float v, float* red, int tid) {
    red[tid] = v; __syncthreads();
    for (int s = 128; s > 0; s >>= 1) {
        if (tid < s) red[tid] = fmaxf(red[tid], red[tid + s]);
        __syncthreads();
    }
    float r = red[0]; __syncthreads();
    return r;
}

__device__ __forceinline__ float blk_reduce_sum(float v, float* red, int tid) {
    red[tid] = v; __syncthreads();
    for (int s = 128; s > 0; s >>= 1) {
        if (tid < s) red[tid] += red[tid + s];
        __syncthreads();
    }
    float r = red[0]; __syncthreads();
    return r;
}

// ---------------- kernel 1: softmaxes, CE rows, ||V_b||^2 ----------------
__global__ __launch_bounds__(256)
void prep_kernel(const float* __restrict__ stu, const float* __restrict__ tea,
                 const int* __restrict__ labels,
                 float* __restrict__ T, float* __restrict__ U,
                 float* __restrict__ Rdiag, float* __restrict__ ce_row) {
    __shared__ float red[256];
    const int b = blockIdx.x;
    const int tid = threadIdx.x;
    const float* trow = tea + (size_t)b * CC;
    const float* srow = stu + (size_t)b * CC;

    // teacher: softmax(t/4)
    float m = -INFINITY;
    for (int c = tid; c < CC; c += 256) m = fmaxf(m, trow[c] * TEMP_INV);
    const float mt = blk_reduce_max(m, red, tid);
    float sum = 0.f;
    for (int c = tid; c < CC; c += 256) sum += expf(trow[c] * TEMP_INV - mt);
    const float st = blk_reduce_sum(sum, red, tid);
    const float rst = 1.0f / st;
    for (int c = tid; c < CC; c += 256)
        T[(size_t)b * CC + c] = expf(trow[c] * TEMP_INV - mt) * rst;

    // student: log_softmax(s/4)
    m = -INFINITY;
    for (int c = tid; c < CC; c += 256) m = fmaxf(m, srow[c] * TEMP_INV);
    const float mu = blk_reduce_max(m, red, tid);
    sum = 0.f;
    for (int c = tid; c < CC; c += 256) sum += expf(srow[c] * TEMP_INV - mu);
    const float su = blk_reduce_sum(sum, red, tid);
    const float lsu = logf(su);
    for (int c = tid; c < CC; c += 256)
        U[(size_t)b * CC + c] = srow[c] * TEMP_INV - mu - lsu;

    // raw student: CE term and squared norm
    m = -INFINITY;
    float sq = 0.f;
    for (int c = tid; c < CC; c += 256) {
        float z = srow[c];
        m = fmaxf(m, z);
        sq += z * z;
    }
    const float mz = blk_reduce_max(m, red, tid);
    sum = 0.f;
    for (int c = tid; c < CC; c += 256) sum += expf(srow[c] - mz);
    const float sz = blk_reduce_sum(sum, red, tid);
    const float sqn = blk_reduce_sum(sq, red, tid);
    if (tid == 0) {
        Rdiag[b] = sqn;
        int lab = labels[b];
        ce_row[b] = mz + logf(sz) - srow[lab];
    }
}

// ---------------- kernel 2: Gram matrix C[i,j] = <Arow_i, Brow_j> via WMMA f32 ----------------
// One wave32 per 16x16 output tile. K swept 4 at a time with V_WMMA_F32_16X16X4_F32.
// f32 16x4 A-fragment layout (ISA 7.12.2): lanes 0-15 -> M=lane, VGPR0/1 = K0/K1;
// lanes 16-31 -> M=lane-16, VGPR0/1 = K2/K3.  B fragment gathers the j-tile rows identically.
__global__ __launch_bounds__(32)
void gram_wmma_kernel(const float* __restrict__ Amat, const float* __restrict__ Bmat,
                      float* __restrict__ Cmat) {
    const int lane  = threadIdx.x;
    const int r     = lane & 15;
    const int khalf = lane >> 4;        // 0 or 1
    const int kof   = khalf << 1;       // 0 or 2
    const int i0 = blockIdx.x * 16;
    const int j0 = blockIdx.y * 16;

    const float* aptr = Amat + (size_t)(i0 + r) * CC + kof;
    const float* bptr = Bmat + (size_t)(j0 + r) * CC + kof;

    v8f acc = {};
#pragma unroll 5
    for (int k = 0; k < CC; k += 4) {
        v2f a = *(const v2f*)(aptr + k);
        v2f bq = *(const v2f*)(bptr + k);
        acc = __builtin_amdgcn_wmma_f32_16x16x4_f32(
            /*neg_a=*/false, a, /*neg_b=*/false, bq,
            /*c_mod=*/(short)0, acc, /*reuse_a=*/false, /*reuse_b=*/false);
    }

    // C/D layout: VGPR t, lanes0-15 -> M=t, lanes16-31 -> M=t+8; N = lane&15
    const int irow = i0 + 8 * khalf;
    const int jj = j0 + r;
#pragma unroll
    for (int t = 0; t < 8; ++t)
        Cmat[(size_t)(irow + t) * BB + jj] = acc[t];
}

// ---------------- kernel 3: inverse norms + Q transpose ----------------
__global__ __launch_bounds__(256)
void norms_kernel(const float* __restrict__ S, const float* __restrict__ P,
                  const float* __restrict__ Q, const float* __restrict__ Rdiag,
                  float* __restrict__ invt, float* __restrict__ invs,
                  float* __restrict__ Qt) {
    const int b = blockIdx.x, i = threadIdx.x;
    const float Sbb = S[(size_t)b * BB + b];
    const float Sib = S[(size_t)b * BB + i];
    const float Sii = S[(size_t)i * BB + i];
    const float nt2 = fmaxf(Sii - 2.f * Sib + Sbb, 0.f);
    invt[(size_t)b * BB + i] = 1.f / fmaxf(sqrtf(nt2), EPSN);

    const float Qib = Q[(size_t)i * BB + b];
    const float Pii = P[(size_t)i * BB + i];
    const float ns2 = fmaxf(Pii - 2.f * Qib + Rdiag[b], 0.f);
    invs[(size_t)b * BB + i] = 1.f / fmaxf(sqrtf(ns2), EPSN);

    Qt[(size_t)b * BB + i] = Qib;
}

// ---------------- kernel 4: B^3 smooth-L1 accumulation, one block per anchor b ----------------
__global__ __launch_bounds__(256)
void rkd_kernel(const float* __restrict__ S, const float* __restrict__ P,
                const float* __restrict__ Qt,
                const float* __restrict__ invt, const float* __restrict__ invs,
                const float* __restrict__ Rdiag, float* __restrict__ partial) {
    __shared__ float sSb[256], sQt[256], sIt[256], sIs[256];
    __shared__ float red[256];
    const int b = blockIdx.x, j = threadIdx.x;

    sSb[j] = S[(size_t)b * BB + j];
    sQt[j] = Qt[(size_t)b * BB + j];
    sIt[j] = invt[(size_t)b * BB + j];
    sIs[j] = invs[(size_t)b * BB + j];
    __syncthreads();

    const float Sbb = sSb[b];
    const float Rbb = Rdiag[b];
    const float Sbj = sSb[j], Qjb = sQt[j], itj = sIt[j], isj = sIs[j];

    float acc = 0.f;
    for (int i = 0; i < BB; ++i) {
        const float tv = (S[(size_t)i * BB + j] - sSb[i] - Sbj + Sbb) * (sIt[i] * itj);
        const float sv = (P[(size_t)i * BB + j] - sQt[i] - Qjb + Rbb) * (sIs[i] * isj);
        const float d = sv - tv;
        const float ad = fabsf(d);
        acc += (ad < 1.f) ? 0.5f * d * d : (ad - 0.5f);
    }

    red[j] = acc; __syncthreads();
    for (int s = 128; s > 0; s >>= 1) {
        if (j < s) red[j] += red[j + s];
        __syncthreads();
    }
    if (j == 0) partial[b] = red[0];
}

// ---------------- kernel 5: final deterministic reduce ----------------
__global__ __launch_bounds__(256)
void final_kernel(const float* __restrict__ partial, const float* __restrict__ ce_row,
                  float* __restrict__ out) {
    __shared__ float r1[256], r2[256];
    const int t = threadIdx.x;
    r1[t] = partial[t];
    r2[t] = ce_row[t];
    __syncthreads();
    for (int s = 128; s > 0; s >>= 1) {
        if (t < s) { r1[t] += r1[t + s]; r2[t] += r2[t + s]; }
        __syncthreads();
    }
    if (t == 0) {
        const float rkd = r1[0] / 16777216.0f;   // B^3
        const float ce  = r2[0] / (float)BB;
        out[0] = rkd * LAMBDA_ + ce;
    }
}

extern "C" void kernel_launch(void* const* d_in, const int* in_sizes, int n_in,
                              void* d_out, int out_size, void* d_ws, size_t ws_size,
                              hipStream_t stream) {
    (void)in_sizes; (void)n_in; (void)out_size; (void)ws_size;
    const float* stu    = (const float*)d_in[0];  // student_logits [256,1000]
    const float* tea    = (const float*)d_in[1];  // teacher_logits [256,1000]
    const int*   labels = (const int*)d_in[2];    // labels [256]

    float* ws = (float*)d_ws;
    float* T      = ws;             // 256000  soft_teacher
    float* U      = T + BB * CC;    // 256000  soft_student (log-softmax)
    float* S      = U + BB * CC;    // 65536   T T^T
    float* P      = S + BB * BB;    // 65536   U U^T
    float* Q      = P + BB * BB;    // 65536   U V^T
    float* Qt     = Q + BB * BB;    // 65536   Q^T
    float* invt   = Qt + BB * BB;   // 65536
    float* invs   = invt + BB * BB; // 65536
    float* Rdiag  = invs + BB * BB; // 256     ||V_b||^2
    float* ce_row = Rdiag + BB;     // 256
    float* partial = ce_row + BB;   // 256

    prep_kernel<<<BB, 256, 0, stream>>>(stu, tea, labels, T, U, Rdiag, ce_row);

    dim3 g(BB / 16, BB / 16);
    gram_wmma_kernel<<<g, 32, 0, stream>>>(T, T, S);
    gram_wmma_kernel<<<g, 32, 0, stream>>>(U, U, P);
    gram_wmma_kernel<<<g, 32, 0, stream>>>(U, stu, Q);

    norms_kernel<<<BB, 256, 0, stream>>>(S, P, Q, Rdiag, invt, invs, Qt);
    rkd_kernel<<<BB, 256, 0, stream>>>(S, P, Qt, invt, invs, Rdiag, partial);
    final_kernel<<<1, 256, 0, stream>>>(partial, ce_row, (float*)d_out);
}